// EdgeCycleLayer_50869592655493
// MI455X (gfx1250) — compile-verified
//
#include <hip/hip_runtime.h>
#include <hip/hip_bf16.h>

// ---------------------------------------------------------------------------
// Problem constants (match reference)
// ---------------------------------------------------------------------------
#define N_NODES 100000
#define NE      400000
#define NC5     40000
#define NC6     40000
#define R5      200000            // 5*NC5 rows (cycle-5 block first)
#define R6      240000            // 6*NC6 rows
#define NR      440000            // total (cycle,slot) rows

typedef __attribute__((ext_vector_type(2))) float v2f;
typedef __attribute__((ext_vector_type(8))) float v8f;

// Workspace layout (float offsets). First ZERO_FLOATS floats are memset to 0.
#define OFF_NODEE   0UL            // [N_NODES,64] edge->node accum of edge_rep
#define OFF_NS5     6400000UL      // [N_NODES,64]
#define OFF_NS6     12800000UL     // [N_NODES,64]
#define OFF_NB5     19200000UL     // [N_NODES,64]
#define OFF_NB6     25600000UL     // [N_NODES,64]
#define OFF_NSLVL   32000000UL     // [N_NODES,64]
#define OFF_NS2     38400000UL     // [N_NODES,128]
#define OFF_NACC2   51200000UL     // [N_NODES,128]
#define ZERO_FLOATS 64000000UL
#define OFF_CT5     64000000UL     // [NC5,64]
#define OFF_CT6     66560000UL     // [NC6,64]
#define OFF_LIFT    69120000UL     // [NR,128] lift_aggr
#define OFF_LVLE    125440000UL    // [NR,64]  lvl_aggr_edge
#define OFF_AEDGE   153600000UL    // [NE,64]  fused A for W2 GEMM

__device__ __forceinline__ int row_node(int r, const int* __restrict__ c5,
                                        const int* __restrict__ c6) {
  return (r < R5) ? c5[r] : c6[r - R5];
}

// ---------------------------------------------------------------------------
// Scatter / gather kernels (memory-bound, atomic segment sums)
// ---------------------------------------------------------------------------

// node_acc_E[n] += edge_rep[e] for both endpoints of every edge
__global__ void k_edge2node(const float* __restrict__ edge_rep,
                            const int* __restrict__ en,
                            float* __restrict__ nodeE) {
  int i = blockIdx.x * blockDim.x + threadIdx.x;
  if (i >= NE * 64) return;
  int e = i >> 6, c = i & 63;
  float v = edge_rep[i];
  int u = en[2 * e], w = en[2 * e + 1];
  atomicAdd(&nodeE[(size_t)u * 64 + c], v);
  atomicAdd(&nodeE[(size_t)w * 64 + c], v);
}

// ns5/ns6: segment-sum of e2c rows (= node_acc_E gathered at row node) per node
__global__ void k_ns56(const float* __restrict__ nodeE,
                       const int* __restrict__ c5, const int* __restrict__ c6,
                       float* __restrict__ ns5, float* __restrict__ ns6) {
  int i = blockIdx.x * blockDim.x + threadIdx.x;
  if (i >= NR * 64) return;
  int r = i >> 6, c = i & 63;
  if (r < R5) {
    int n = c5[r];
    atomicAdd(&ns5[(size_t)n * 64 + c], nodeE[(size_t)n * 64 + c]);
  } else {
    int n = c6[r - R5];
    atomicAdd(&ns6[(size_t)n * 64 + c], nodeE[(size_t)n * 64 + c]);
  }
}

// per-cycle totals ct[q] = sum over slots of node_acc_E[cyc_nodes[q,s]]
__global__ void k_ct(const float* __restrict__ nodeE,
                     const int* __restrict__ cyc_nodes, int slots, int ncyc,
                     float* __restrict__ ct) {
  int i = blockIdx.x * blockDim.x + threadIdx.x;
  if (i >= ncyc * 64) return;
  int q = i >> 6, c = i & 63;
  float s = 0.f;
  for (int sl = 0; sl < slots; ++sl) {
    int n = cyc_nodes[q * slots + sl];
    s += nodeE[(size_t)n * 64 + c];
  }
  ct[i] = s;
}

// nb[n] += ct[cycle(r)] for every row r of a family
__global__ void k_nb(const float* __restrict__ ct,
                     const int* __restrict__ cyc_nodes, int slots, int nrows,
                     float* __restrict__ nb) {
  int i = blockIdx.x * blockDim.x + threadIdx.x;
  if (i >= nrows * 64) return;
  int r = i >> 6, c = i & 63;
  int q = r / slots;
  int n = cyc_nodes[r];
  atomicAdd(&nb[(size_t)n * 64 + c], ct[(size_t)q * 64 + c]);
}

// lift_aggr[r] = [ns(row(r)) | nb(row(r))] per family
__global__ void k_lift(const float* __restrict__ ns5, const float* __restrict__ nb5,
                       const float* __restrict__ ns6, const float* __restrict__ nb6,
                       const int* __restrict__ c5, const int* __restrict__ c6,
                       float* __restrict__ lift) {
  int i = blockIdx.x * blockDim.x + threadIdx.x;
  if (i >= NR * 128) return;
  int r = i >> 7, c = i & 127;
  float v;
  if (r < R5) {
    int n = c5[r];
    v = (c < 64) ? ns5[(size_t)n * 64 + c] : nb5[(size_t)n * 64 + (c - 64)];
  } else {
    int n = c6[r - R5];
    v = (c < 64) ? ns6[(size_t)n * 64 + c] : nb6[(size_t)n * 64 + (c - 64)];
  }
  lift[i] = v;
}

// ns_lvl[n] += lvl_aggr_edge[r] over all rows
__global__ void k_nslvl(const float* __restrict__ lvle,
                        const int* __restrict__ c5, const int* __restrict__ c6,
                        float* __restrict__ nslvl) {
  int i = blockIdx.x * blockDim.x + threadIdx.x;
  if (i >= NR * 64) return;
  int r = i >> 6, c = i & 63;
  int n = row_node(r, c5, c6);
  atomicAdd(&nslvl[(size_t)n * 64 + c], lvle[i]);
}

// ns2[n] += cycle_rep[r] over all rows
__global__ void k_ns2(const float* __restrict__ cyc_rep,
                      const int* __restrict__ c5, const int* __restrict__ c6,
                      float* __restrict__ ns2) {
  int i = blockIdx.x * blockDim.x + threadIdx.x;
  if (i >= NR * 128) return;
  int r = i >> 7, c = i & 127;
  int n = row_node(r, c5, c6);
  atomicAdd(&ns2[(size_t)n * 128 + c], cyc_rep[i]);
}

// A_edge[e] = (1+eps1_1)*edge_rep[e] + (1+eps1_2)*(ns_lvl[u]+ns_lvl[v])
__global__ void k_aedge(const float* __restrict__ edge_rep,
                        const float* __restrict__ nslvl,
                        const int* __restrict__ en,
                        const float* __restrict__ eps11,
                        const float* __restrict__ eps12,
                        float* __restrict__ Aedge) {
  int i = blockIdx.x * blockDim.x + threadIdx.x;
  if (i >= NE * 64) return;
  int e = i >> 6, c = i & 63;
  int u = en[2 * e], w = en[2 * e + 1];
  float s1 = 1.f + eps11[0], s2 = 1.f + eps12[0];
  Aedge[i] = s1 * edge_rep[i] +
             s2 * (nslvl[(size_t)u * 64 + c] + nslvl[(size_t)w * 64 + c]);
}

// node_acc2[u] += ns2[u]+ns2[v]; node_acc2[v] += ns2[u]+ns2[v] per edge
__global__ void k_nacc2(const float* __restrict__ ns2,
                        const int* __restrict__ en,
                        float* __restrict__ nacc2) {
  int i = blockIdx.x * blockDim.x + threadIdx.x;
  if (i >= NE * 128) return;
  int e = i >> 7, c = i & 127;
  int u = en[2 * e], w = en[2 * e + 1];
  float t = ns2[(size_t)u * 128 + c] + ns2[(size_t)w * 128 + c];
  atomicAdd(&nacc2[(size_t)u * 128 + c], t);
  atomicAdd(&nacc2[(size_t)w * 128 + c], t);
}

// ---------------------------------------------------------------------------
// WMMA fp32 GEMM kernels: D = relu(A @ W), V_WMMA_F32_16X16X4_F32
// Each wave owns a 16x16 tile; A tile staged in LDS, W staged once per block.
// ---------------------------------------------------------------------------

// GEMM1: [NR,256] ([lift | cycle_rep]) @ W1[256,64] -> lvl_aggr_edge [NR,64]
__global__ void gemm1(const float* __restrict__ lift,
                      const float* __restrict__ cyc,
                      const float* __restrict__ W,    // [256,64]
                      float* __restrict__ out, int mtiles) {
  constexpr int K = 256, N = 64, AP = K + 4, WP = N + 4;
  extern __shared__ float smem[];
  float* As = smem;           // 16*AP
  float* Ws = smem + 16 * AP; // K*WP
  const int tid = threadIdx.x;
  for (int i = tid; i < K * N; i += blockDim.x) {
    int k = i / N, n = i % N;
    Ws[k * WP + n] = W[i];
  }
  const int lane = tid & 31, wv = tid >> 5;
  const int m = lane & 15, kh = (lane >> 4) << 1, n0 = wv * 16;
  const int mup = (lane >> 4) << 3;  // +8 rows for upper half-wave on store
  for (int t = blockIdx.x; t < mtiles; t += gridDim.x) {
    const int r0 = t * 16;
    __syncthreads();
    for (int i = tid; i < 16 * K; i += blockDim.x) {
      int j = i / K, k = i % K;
      int r = r0 + j;
      As[j * AP + k] = (k < 128) ? lift[(size_t)r * 128 + k]
                                 : cyc[(size_t)r * 128 + (k - 128)];
    }
    __syncthreads();
    v8f acc = {0.f, 0.f, 0.f, 0.f, 0.f, 0.f, 0.f, 0.f};
#pragma unroll 8
    for (int k0 = 0; k0 < K; k0 += 4) {
      v2f a, b;
      a.x = As[m * AP + k0 + kh];
      a.y = As[m * AP + k0 + kh + 1];
      b.x = Ws[(k0 + kh) * WP + n0 + m];
      b.y = Ws[(k0 + kh + 1) * WP + n0 + m];
      acc = __builtin_amdgcn_wmma_f32_16x16x4_f32(false, a, false, b,
                                                  (short)0, acc, false, false);
    }
#pragma unroll
    for (int v = 0; v < 8; ++v) {
      float val = acc[v];
      out[(size_t)(r0 + v + mup) * N + n0 + m] = val > 0.f ? val : 0.f;
    }
  }
}

// GEMM2: A_edge[NE,64] @ W2[64,64] -> edge_out [NE,64]
__global__ void gemm2(const float* __restrict__ A,
                      const float* __restrict__ W,    // [64,64]
                      float* __restrict__ out, int mtiles) {
  constexpr int K = 64, N = 64, AP = K + 4, WP = N + 4;
  extern __shared__ float smem[];
  float* As = smem;
  float* Ws = smem + 16 * AP;
  const int tid = threadIdx.x;
  for (int i = tid; i < K * N; i += blockDim.x) {
    int k = i / N, n = i % N;
    Ws[k * WP + n] = W[i];
  }
  const int lane = tid & 31, wv = tid >> 5;
  const int m = lane & 15, kh = (lane >> 4) << 1, n0 = wv * 16;
  const int mup = (lane >> 4) << 3;
  for (int t = blockIdx.x; t < mtiles; t += gridDim.x) {
    const int r0 = t * 16;
    __syncthreads();
    for (int i = tid; i < 16 * K; i += blockDim.x) {
      int j = i / K, k = i % K;
      As[j * AP + k] = A[(size_t)(r0 + j) * K + k];
    }
    __syncthreads();
    v8f acc = {0.f, 0.f, 0.f, 0.f, 0.f, 0.f, 0.f, 0.f};
#pragma unroll 8
    for (int k0 = 0; k0 < K; k0 += 4) {
      v2f a, b;
      a.x = As[m * AP + k0 + kh];
      a.y = As[m * AP + k0 + kh + 1];
      b.x = Ws[(k0 + kh) * WP + n0 + m];
      b.y = Ws[(k0 + kh + 1) * WP + n0 + m];
      acc = __builtin_amdgcn_wmma_f32_16x16x4_f32(false, a, false, b,
                                                  (short)0, acc, false, false);
    }
#pragma unroll
    for (int v = 0; v < 8; ++v) {
      float val = acc[v];
      out[(size_t)(r0 + v + mup) * N + n0 + m] = val > 0.f ? val : 0.f;
    }
  }
}

// GEMM3: A[r] = (1+eps2)*node_acc2[row(r)] + lift[r]  (fused gather),
//        A[NR,128] @ W_lift[128,128] -> cycle_out [NR,128]
__global__ void gemm3(const float* __restrict__ nacc2,
                      const float* __restrict__ lift,
                      const float* __restrict__ W,    // [128,128]
                      const int* __restrict__ c5, const int* __restrict__ c6,
                      const float* __restrict__ eps2p,
                      float* __restrict__ out, int mtiles) {
  constexpr int K = 128, N = 128, AP = K + 4, WP = N + 4;
  extern __shared__ float smem[];
  float* As = smem;           // 16*AP
  float* Ws = smem + 16 * AP; // K*WP
  const int tid = threadIdx.x;
  const float s2 = 1.f + eps2p[0];
  for (int i = tid; i < K * N; i += blockDim.x) {
    int k = i / N, n = i % N;
    Ws[k * WP + n] = W[i];
  }
  const int lane = tid & 31, wv = tid >> 5;  // 8 waves
  const int m = lane & 15, kh = (lane >> 4) << 1, n0 = wv * 16;
  const int mup = (lane >> 4) << 3;
  for (int t = blockIdx.x; t < mtiles; t += gridDim.x) {
    const int r0 = t * 16;
    __syncthreads();
    for (int i = tid; i < 16 * K; i += blockDim.x) {
      int j = i / K, k = i % K;
      int r = r0 + j;
      int n = row_node(r, c5, c6);
      As[j * AP + k] = s2 * nacc2[(size_t)n * 128 + k] + lift[(size_t)r * 128 + k];
    }
    __syncthreads();
    v8f acc = {0.f, 0.f, 0.f, 0.f, 0.f, 0.f, 0.f, 0.f};
#pragma unroll 8
    for (int k0 = 0; k0 < K; k0 += 4) {
      v2f a, b;
      a.x = As[m * AP + k0 + kh];
      a.y = As[m * AP + k0 + kh + 1];
      b.x = Ws[(k0 + kh) * WP + n0 + m];
      b.y = Ws[(k0 + kh + 1) * WP + n0 + m];
      acc = __builtin_amdgcn_wmma_f32_16x16x4_f32(false, a, false, b,
                                                  (short)0, acc, false, false);
    }
#pragma unroll
    for (int v = 0; v < 8; ++v) {
      float val = acc[v];
      out[(size_t)(r0 + v + mup) * N + n0 + m] = val > 0.f ? val : 0.f;
    }
  }
}

// ---------------------------------------------------------------------------
// Host launcher
// ---------------------------------------------------------------------------
static inline int nblk(long total, int tpb) { return (int)((total + tpb - 1) / tpb); }

extern "C" void kernel_launch(void* const* d_in, const int* in_sizes, int n_in,
                              void* d_out, int out_size, void* d_ws, size_t ws_size,
                              hipStream_t stream) {
  (void)in_sizes; (void)n_in; (void)out_size; (void)ws_size;
  const float* edge_rep  = (const float*)d_in[0];
  const float* cycle_rep = (const float*)d_in[1];
  const float* W_lift    = (const float*)d_in[2];
  const float* W1        = (const float*)d_in[3];
  const float* W2        = (const float*)d_in[4];
  const float* eps1_1    = (const float*)d_in[5];
  const float* eps1_2    = (const float*)d_in[6];
  const float* eps2      = (const float*)d_in[7];
  const int*   edge_nodes = (const int*)d_in[8];
  const int*   c5 = (const int*)d_in[9];
  const int*   c6 = (const int*)d_in[10];

  float* out_edge = (float*)d_out;                       // [NE,64]
  float* out_cyc  = out_edge + (size_t)NE * 64;          // [NR,128]

  float* ws     = (float*)d_ws;
  float* nodeE  = ws + OFF_NODEE;
  float* ns5    = ws + OFF_NS5;
  float* ns6    = ws + OFF_NS6;
  float* nb5    = ws + OFF_NB5;
  float* nb6    = ws + OFF_NB6;
  float* nslvl  = ws + OFF_NSLVL;
  float* ns2    = ws + OFF_NS2;
  float* nacc2  = ws + OFF_NACC2;
  float* ct5    = ws + OFF_CT5;
  float* ct6    = ws + OFF_CT6;
  float* lift   = ws + OFF_LIFT;
  float* lvle   = ws + OFF_LVLE;
  float* Aedge  = ws + OFF_AEDGE;

  const int TPB = 256;

  // zero all atomic accumulators in one memset (capture-safe)
  hipMemsetAsync(ws, 0, ZERO_FLOATS * sizeof(float), stream);

  // edge -> node accumulation of edge_rep
  k_edge2node<<<nblk((long)NE * 64, TPB), TPB, 0, stream>>>(edge_rep, edge_nodes, nodeE);
  // per-family node segment sums of gathered e2c rows
  k_ns56<<<nblk((long)NR * 64, TPB), TPB, 0, stream>>>(nodeE, c5, c6, ns5, ns6);
  // per-cycle totals
  k_ct<<<nblk((long)NC5 * 64, TPB), TPB, 0, stream>>>(nodeE, c5, 5, NC5, ct5);
  k_ct<<<nblk((long)NC6 * 64, TPB), TPB, 0, stream>>>(nodeE, c6, 6, NC6, ct6);
  // broadcast cycle totals back to nodes
  k_nb<<<nblk((long)R5 * 64, TPB), TPB, 0, stream>>>(ct5, c5, 5, R5, nb5);
  k_nb<<<nblk((long)R6 * 64, TPB), TPB, 0, stream>>>(ct6, c6, 6, R6, nb6);
  // lift_aggr rows
  k_lift<<<nblk((long)NR * 128, TPB), TPB, 0, stream>>>(ns5, nb5, ns6, nb6, c5, c6, lift);

  // GEMM1: relu([lift|cycle_rep] @ W1) -> lvl_aggr_edge
  {
    size_t shmem = (16 * (256 + 4) + 256 * (64 + 4)) * sizeof(float);
    gemm1<<<2048, 128, shmem, stream>>>(lift, cycle_rep, W1, lvle, NR / 16);
  }

  // node segment sums for second stage
  k_nslvl<<<nblk((long)NR * 64, TPB), TPB, 0, stream>>>(lvle, c5, c6, nslvl);
  k_ns2<<<nblk((long)NR * 128, TPB), TPB, 0, stream>>>(cycle_rep, c5, c6, ns2);

  // fused A for the edge GEMM, and node_acc2 for the cycle GEMM
  k_aedge<<<nblk((long)NE * 64, TPB), TPB, 0, stream>>>(edge_rep, nslvl, edge_nodes,
                                                        eps1_1, eps1_2, Aedge);
  k_nacc2<<<nblk((long)NE * 128, TPB), TPB, 0, stream>>>(ns2, edge_nodes, nacc2);

  // GEMM2: relu(A_edge @ W2) -> edge_out
  {
    size_t shmem = (16 * (64 + 4) + 64 * (64 + 4)) * sizeof(float);
    gemm2<<<2048, 128, shmem, stream>>>(Aedge, W2, out_edge, NE / 16);
  }
  // GEMM3: relu(((1+eps2)*linmap + lift) @ W_lift) -> cycle_out (A gathered on the fly)
  {
    size_t shmem = (16 * (128 + 4) + 128 * (128 + 4)) * sizeof(float);
    gemm3<<<2048, 256, shmem, stream>>>(nacc2, lift, W_lift, c5, c6, eps2,
                                        out_cyc, NR / 16);
  }
}